// PairingLayer_82162724373211
// MI455X (gfx1250) — compile-verified
//
#include <hip/hip_runtime.h>
#include <hip/hip_bf16.h>

typedef __attribute__((ext_vector_type(2))) float v2f;
typedef __attribute__((ext_vector_type(8))) float v8f;

#define B_DIM 8
#define N_DIM 128
#define D_DIM 256
// w layout: [0:256)=w_m, [256:512)=w_mT, [512:768)=w_prod, [768:1023)=w_oh (K=255)

// ---------------------------------------------------------------------------
// Single fused kernel.
// Grid: 64 blocks = (b, i_tile); block = 256 threads = 8 waves; wave = j-tile.
//  1. Async-copy the 16x256 A panel (rows ibase..ibase+15 of X[b]) to LDS
//     with global_load_async_to_lds_b128 (ASYNCcnt), shared by all 8 waves.
//  2. Each wave runs the K=256 fp32 WMMA loop (V_WMMA_F32_16X16X4_F32,
//     K-step 4): A fragments from LDS, B fragments + w_prod from global
//     (L2-resident). The row/col linear terms r[i], c[j] are accumulated in
//     the same loop (2 extra FMAs/step) and wave-reduced with shfl_xor(16).
//  3. Epilogue adds r + c + one-hot gather + bias, applies the validity mask,
//     writes logits, then a block-level argmax over the 16x128 LDS slab
//     produces preds (first-occurrence tie-break, all -inf row -> 0).
// ---------------------------------------------------------------------------
__global__ __launch_bounds__(256) void pair_fused_kernel(
    const float* __restrict__ x,      // [8,128,256]
    const int* __restrict__ um,       // unit_mask [8,128]
    const float* __restrict__ w,      // [1023]
    const float* __restrict__ bias_p, // scalar
    float* __restrict__ out) {        // 131072 logits + 1024 preds (as float)
  __shared__ float aS[16 * D_DIM];        // 16 KB: A panel (raw X rows)
  __shared__ float logits_s[16 * N_DIM];  // 8 KB: logit slab for argmax

  const int b     = (int)(blockIdx.x >> 3);
  const int itile = (int)(blockIdx.x & 7);
  const int wv    = (int)(threadIdx.x >> 5);   // j tile (0..7)
  const int lane  = (int)(threadIdx.x & 31);
  const int half  = lane >> 4;                 // K-half select
  const int sub   = lane & 15;
  const int ibase = itile * 16;
  const int jbase = wv * 16;

  const float* xb     = x + (size_t)b * N_DIM * D_DIM;
  const float* xj_row = xb + (size_t)(jbase + sub) * D_DIM;  // B row j = jbase+sub
  __builtin_prefetch(xj_row, 0, 0);

  // ---- stage A panel (16 rows x 256 f32 = 16 KB, contiguous) into LDS ----
  {
    const unsigned lds_base = (unsigned)(unsigned long long)(&aS[0]);
    const char* gsrc = (const char*)(xb + (size_t)ibase * D_DIM);
#pragma unroll
    for (int q = 0; q < 4; ++q) {
      const unsigned off  = ((unsigned)threadIdx.x + q * 256u) * 16u;
      const unsigned ldsa = lds_base + off;
      const char*    ga   = gsrc + off;
      asm volatile("global_load_async_to_lds_b128 %0, %1, off"
                   :: "v"(ldsa), "v"(ga) : "memory");
    }
    asm volatile("s_wait_asynccnt 0x0" ::: "memory");
  }
  __syncthreads();

  // ---- fused WMMA + linear-term loop over K ----
  // A(16x4 f32): lanes 0-15 hold M=sub with K={kk,kk+1}; lanes 16-31 K={kk+2,kk+3}.
  // B(4x16) mirrors with N striped across lanes. C/D: v8f, M = v + 8*half, N = sub.
  const float* wp = w + 2 * D_DIM;  // w_prod
  v8f acc = {};
  float rpart = 0.0f, cpart = 0.0f;
#pragma unroll 8
  for (int kk = 0; kk < D_DIM; kk += 4) {
    const int k0 = kk + 2 * half;
    const float2 av  = *(const float2*)&aS[sub * D_DIM + k0];  // ds_load_b64
    const float2 bv  = *(const float2*)(xj_row + k0);
    const float2 wp2 = *(const float2*)(wp + k0);
    const float2 wr2 = *(const float2*)(w + k0);            // w_m
    const float2 wc2 = *(const float2*)(w + D_DIM + k0);    // w_mT
    rpart = fmaf(av.x, wr2.x, rpart);
    rpart = fmaf(av.y, wr2.y, rpart);
    cpart = fmaf(bv.x, wc2.x, cpart);
    cpart = fmaf(bv.y, wc2.y, cpart);
    v2f afrag, bfrag;
    afrag.x = av.x;
    afrag.y = av.y;
    bfrag.x = bv.x * wp2.x;  // fold diag(w_prod) into B
    bfrag.y = bv.y * wp2.y;
    acc = __builtin_amdgcn_wmma_f32_16x16x4_f32(
        /*neg_a=*/false, afrag, /*neg_b=*/false, bfrag,
        /*c_mod=*/(short)0, acc, /*reuse_a=*/false, /*reuse_b=*/false);
  }
  // combine the two K-halves: lanes l and l+16 hold the same row/col
  const float rval = rpart + __shfl_xor(rpart, 16, 32);  // r[ibase+sub]
  const float cval = cpart + __shfl_xor(cpart, 16, 32);  // c[jbase+sub]

  // ---- epilogue: r + c + one-hot + bias, mask, write logits ----
  const float biasv = bias_p[0];
  const int j = jbase + sub;
  const bool mj = um[b * N_DIM + j] != 0;
  float* outb = out + ((size_t)b * N_DIM + ibase) * N_DIM;
  const float NEG_INF = -__builtin_inff();

#pragma unroll
  for (int v = 0; v < 8; ++v) {
    const int mrow = v + 8 * half;  // row within tile (C/D layout)
    const int i = ibase + mrow;
    const bool mi = um[b * N_DIM + i] != 0;
    const float ri = __shfl(rval, mrow, 32);  // r for row mrow lives in lane mrow
    const int a_val = (b * N_DIM + i) >> 3;   // (b*N+i)//B
    const float oh = w[3 * D_DIM + (j + (N_DIM - 1) - a_val)];  // in [0,255)
    const float val = acc[v] + ri + cval + oh + biasv;
    const float lg = (mi && mj) ? val : NEG_INF;
    outb[(size_t)mrow * N_DIM + j] = lg;
    logits_s[mrow * N_DIM + j] = lg;
  }
  __syncthreads();

  // ---- fused argmax over j for the block's 16 rows (2 rows per wave) ----
#pragma unroll
  for (int rr = 0; rr < 2; ++rr) {
    const int mrow = 2 * wv + rr;
    float bestv = NEG_INF;
    int bestj = 0;  // all -inf row -> 0, matching jnp.argmax
#pragma unroll
    for (int t = 0; t < 4; ++t) {
      const int jj = lane + 32 * t;
      const float vv = logits_s[mrow * N_DIM + jj];
      if (vv > bestv) { bestv = vv; bestj = jj; }  // strict > keeps first occurrence
    }
#pragma unroll
    for (int off = 16; off >= 1; off >>= 1) {
      const float ov = __shfl_xor(bestv, off, 32);
      const int   oj = __shfl_xor(bestj, off, 32);
      if (ov > bestv || (ov == bestv && oj < bestj)) { bestv = ov; bestj = oj; }
    }
    if (lane == 0) {
      out[(size_t)B_DIM * N_DIM * N_DIM + (b * N_DIM + ibase + mrow)] = (float)bestj;
    }
  }
}

// ---------------------------------------------------------------------------
extern "C" void kernel_launch(void* const* d_in, const int* in_sizes, int n_in,
                              void* d_out, int out_size, void* d_ws, size_t ws_size,
                              hipStream_t stream) {
  const float* x    = (const float*)d_in[0];  // [8,128,256] f32
  const int*   um   = (const int*)d_in[1];    // [8,128] mask
  const float* w    = (const float*)d_in[2];  // [1023] f32
  const float* bias = (const float*)d_in[3];  // scalar f32
  float* out = (float*)d_out;                 // 131072 logits + 1024 preds

  // single launch: 64 blocks = (b, i_tile), 8 waves each = j tiles
  pair_fused_kernel<<<64, 256, 0, stream>>>(x, um, w, bias, out);
}